// ReaRev_62388694942425
// MI455X (gfx1250) — compile-verified
//
#include <hip/hip_runtime.h>
#include <cmath>

// ---------------- dimensions (match reference) ----------------
#define NN     20000
#define EE     100000
#define BB     16
#define SS     32
#define HH     128
#define KK     2
#define LL     2
#define NITER  2
#define NODE_IN 256
#define EDGE_DIM 128
#define QDIM   768

typedef __attribute__((ext_vector_type(2))) float v2f;
typedef __attribute__((ext_vector_type(8))) float v8f;
typedef __attribute__((ext_vector_type(4))) unsigned int v4u;
typedef __attribute__((ext_vector_type(4))) int v4i;
typedef __attribute__((ext_vector_type(8))) int v8i;

#if defined(__HIP_DEVICE_COMPILE__) && __has_builtin(__builtin_amdgcn_tensor_load_to_lds)
#define HAVE_TDM 1
#else
#define HAVE_TDM 0
#endif

// =================================================================
// TDM: load a 64x32 f32 tile (row stride = src_stride elements) into LDS,
// padding +2 dwords after every 32 dwords (-> LDS row stride 34), rows
// beyond rows_valid zero-filled via tensor OOB semantics.
// D# layout per CDNA5 ISA 8.3/8.4.
// =================================================================
#if HAVE_TDM
__device__ __forceinline__ void tdm_load_tile_64x32(unsigned lds_byte_off,
                                                    const float* gsrc,
                                                    int rows_valid,
                                                    int src_stride) {
    unsigned long long ga = (unsigned long long)(uintptr_t)gsrc;
    v4u g0;
    g0[0] = 1u;                                   // count=1, is_restore=0, gather=0
    g0[1] = lds_byte_off;                         // lds_addr
    g0[2] = (unsigned)ga;                         // global_addr[31:0]
    g0[3] = (unsigned)((ga >> 32) & 0x01FFFFFFu)  // global_addr[56:32]
          | (2u << 30);                           // type=2 ("image")
    v8i g1;
    g1[0] = (int)((2u << 16)                      // data_size = 4B
                | (1u << 20)                      // pad_enable
                | (4u << 22)                      // pad_interval: 32 dwords
                | (1u << 25));                    // pad_amount: 2 dwords
    g1[1] = (int)(32u << 16);                     // tensor_dim0 = 32 (cols)
    g1[2] = (int)((unsigned)rows_valid << 16);    // tensor_dim1 low16 (rows valid)
    g1[3] = (int)(32u << 16);                     // tile_dim0 = 32
    g1[4] = 64;                                   // tile_dim1 = 64, tile_dim2 = 0
    g1[5] = src_stride;                           // tensor_dim0_stride (elements)
    g1[6] = 0;
    g1[7] = 0;
    v4i z4 = {0, 0, 0, 0};
#if __clang_major__ >= 23
    v8i z8 = {0, 0, 0, 0, 0, 0, 0, 0};
    __builtin_amdgcn_tensor_load_to_lds(g0, g1, z4, z4, z8, 0);
#else
    __builtin_amdgcn_tensor_load_to_lds(g0, g1, z4, z4, 0);
#endif
}
#endif

// =================================================================
// WMMA-f32 GEMM:  C[M,128] = act( Acat[M,Ka] @ W[Ka,128] + bias )
// A virtual concat: cols [0,K0) from A0 (ld0), cols [K0,Ka) from A1 (ld1);
// every 32-wide K chunk lies entirely inside one source (K0 % 32 == 0).
// Block 256 threads (8 waves); tile 64x128; wave -> 16x64 strip.
// A tile staged by the Tensor Data Mover (wave 0) while all threads restage
// the W chunk into a paired-k layout so B fragments are single ds_load_b64.
// =================================================================
#define ASTR 34    // A tile LDS row stride (dwords) == TDM pad layout
#define WPS  288   // W paired-k row stride: 128 cols * 2 + 32 pad (bank split)

__global__ void gemm_wmma_f32(const float* __restrict__ A0, int ld0, int K0,
                              const float* A1, int ld1,
                              const float* __restrict__ W,     // [Ka,128] row-major
                              const float* __restrict__ bias,  // [128]
                              float* __restrict__ C,           // [M,128]
                              int M, int Ka, int do_relu) {
    __shared__ float As[64 * ASTR];
    __shared__ float Wp[16 * WPS];

    const int tid   = threadIdx.x;
    const int wave  = tid >> 5;
    const int lane  = tid & 31;
    const int half  = lane >> 4;   // lanes 0-15 vs 16-31
    const int l16   = lane & 15;
    const int mt    = wave & 3;           // m sub-tile 0..3
    const int ntB   = (wave >> 2) * 64;   // n strip base: 0 or 64
    const int mBase = blockIdx.x * 64;
    const int rowsValid = min(64, M - mBase);

    v8f acc[4];
#pragma unroll
    for (int j = 0; j < 4; ++j)
#pragma unroll
        for (int r = 0; r < 8; ++r) acc[j][r] = 0.0f;

    for (int k0 = 0; k0 < Ka; k0 += 32) {
        // ---- A chunk source (uniform per chunk; concat split is 32-aligned) ----
        const float* srcA;
        int ldc;
        if (k0 < K0) { srcA = A0 + (size_t)mBase * ld0 + k0;        ldc = ld0; }
        else         { srcA = A1 + (size_t)mBase * ld1 + (k0 - K0); ldc = ld1; }

#if HAVE_TDM
        if (wave == 0)
            tdm_load_tile_64x32((unsigned)(uintptr_t)&As[0], srcA, rowsValid, ldc);
#else
        for (int idx = tid; idx < 64 * 32; idx += 256) {
            int r = idx >> 5, c = idx & 31;
            As[r * ASTR + c] = (r < rowsValid) ? srcA[(size_t)r * ldc + c] : 0.0f;
        }
#endif
        // ---- W chunk -> paired-k layout: Wp[k>>1][col*2 + (k&1)] ----
        for (int idx = tid; idx < 32 * 128; idx += 256) {
            int r = idx >> 7, c = idx & 127;
            Wp[(r >> 1) * WPS + c * 2 + (r & 1)] = W[(size_t)(k0 + r) * 128 + c];
        }
#if HAVE_TDM
        if (wave == 0) __builtin_amdgcn_s_wait_tensorcnt(0);
#endif
        __syncthreads();

#pragma unroll
        for (int kk = 0; kk < 32; kk += 4) {
            const int ka = kk + half * 2;  // 16x4-f32 A layout: K split by lane half
            v2f a;
            a.x = As[(mt * 16 + l16) * ASTR + ka];
            a.y = As[(mt * 16 + l16) * ASTR + ka + 1];
            const int kp = (ka >> 1) * WPS;
#pragma unroll
            for (int j = 0; j < 4; ++j) {
                const int col = ntB + j * 16 + l16;
                v2f b = *(const v2f*)&Wp[kp + col * 2];   // one ds_load_b64
                acc[j] = __builtin_amdgcn_wmma_f32_16x16x4_f32(
                    false, a, false, b, (short)0, acc[j], false, false);
            }
        }
        __syncthreads();
    }

    // ---- epilogue: C VGPR r -> rows (r, r+8) across lane halves ----
#pragma unroll
    for (int j = 0; j < 4; ++j) {
        const int col = ntB + j * 16 + l16;
        const float bv = bias ? bias[col] : 0.0f;
#pragma unroll
        for (int r = 0; r < 8; ++r) {
            const int row = mBase + mt * 16 + half * 8 + r;
            if (row < M) {
                float v = acc[j][r] + bv;
                if (do_relu) v = v > 0.0f ? v : 0.0f;
                C[(size_t)row * 128 + col] = v;
            }
        }
    }
}

// =================================================================
// Small kernels
// =================================================================
__global__ void q_repr_kernel(const float* __restrict__ qt, const float* __restrict__ mask,
                              float* __restrict__ qr) {
    const int b = blockIdx.x, j = threadIdx.x;
    float s = 0.0f, ms = 0.0f;
    for (int t = 0; t < SS; ++t) {
        float m = mask[b * SS + t];
        s  += qt[((size_t)(b * SS + t)) * HH + j] * m;
        ms += m;
    }
    qr[b * HH + j] = s / fmaxf(ms, 1e-9f);
}

__global__ void instr_gen_kernel(const float* __restrict__ qt, const float* __restrict__ mask,
                                 const float* __restrict__ qr, const float* __restrict__ instr_init,
                                 const float* __restrict__ Wk, const float* __restrict__ bk,
                                 const float* __restrict__ Wtok, const float* __restrict__ btok,
                                 float* __restrict__ instr) {
    const int b = blockIdx.x, j = threadIdx.x;   // 128 threads
    __shared__ float prev[HH], qrs[HH], qk[HH], sc[SS], wts[SS];
    prev[j] = instr_init[j];
    qrs[j]  = qr[b * HH + j];
    __syncthreads();

    for (int k = 0; k < KK; ++k) {
        const float* Wkk = Wk + (size_t)k * 4 * HH * HH;
        float acc = bk[k * HH + j];
        for (int i = 0; i < HH; ++i) {
            float pv = prev[i], qv = qrs[i];
            acc += pv        * Wkk[(size_t)(i)        * HH + j];
            acc += qv        * Wkk[(size_t)(HH + i)   * HH + j];
            acc += (qv * pv) * Wkk[(size_t)(2*HH + i) * HH + j];
            acc += (qv - pv) * Wkk[(size_t)(3*HH + i) * HH + j];
        }
        qk[j] = acc;
        __syncthreads();

        const int wv = j >> 5, ln = j & 31;
        for (int s = wv; s < SS; s += 4) {
            float a = 0.0f;
            for (int jj = ln; jj < HH; jj += 32)
                a += qk[jj] * qt[((size_t)(b * SS + s)) * HH + jj] * Wtok[jj];
            for (int off = 16; off > 0; off >>= 1) a += __shfl_down(a, off, 32);
            if (ln == 0) {
                a += btok[0];
                if (mask[b * SS + s] == 0.0f) a = -1e9f;
                sc[s] = a;
            }
        }
        __syncthreads();
        if (j == 0) {
            float mx = sc[0];
            for (int s = 1; s < SS; ++s) mx = fmaxf(mx, sc[s]);
            float sm = 0.0f;
            for (int s = 0; s < SS; ++s) { wts[s] = __expf(sc[s] - mx); sm += wts[s]; }
            for (int s = 0; s < SS; ++s) wts[s] /= sm;
        }
        __syncthreads();
        float ik = 0.0f;
        for (int s = 0; s < SS; ++s) ik += wts[s] * qt[(size_t)(b * SS + s) * HH + j];
        instr[(size_t)(b * KK + k) * HH + j] = ik;
        __syncthreads();
        prev[j] = ik;
        __syncthreads();
    }
}

__global__ void seed_kernel(const int* __restrict__ seed, float* __restrict__ p, int n) {
    int i = blockIdx.x * blockDim.x + threadIdx.x;
    if (i < n) p[i] = (float)seed[i];
}

__global__ void edge_message_scatter(const int* __restrict__ src, const int* __restrict__ dst,
                                     const int* __restrict__ etype, const int* __restrict__ batch,
                                     const float* __restrict__ ep, const float* __restrict__ instr,
                                     const float* __restrict__ p, float* __restrict__ agg, int E) {
    const int e = blockIdx.x;
    const int j = threadIdx.x;   // 0..127
    if (e >= E) return;
    const int s = src[e], d = dst[e];
    const int b = batch[d];
    const float pe  = p[s];
    const float epj = ep[(size_t)e * HH + j];
    const int off = (etype[e] == 1) ? KK : 0;   // concat([fwd, inv]) along instruction axis
    float* aggRow = agg + (size_t)d * (2 * KK * HH);
#pragma unroll
    for (int k = 0; k < KK; ++k) {
        float m = instr[(size_t)(b * KK + k) * HH + j] * epj;
        m = m > 0.0f ? m : 0.0f;   // relu
        m *= pe;
        atomicAdd(&aggRow[(off + k) * HH + j], m);
    }
}

__global__ void logits_kernel(const float* __restrict__ h, const float* __restrict__ Wp,
                              const float* __restrict__ bp, float* __restrict__ logits, int n) {
    const int warp = blockIdx.x * (blockDim.x >> 5) + (threadIdx.x >> 5);
    const int lane = threadIdx.x & 31;
    if (warp >= n) return;
    const float* hr = h + (size_t)warp * HH;
    float acc = 0.0f;
    for (int j = lane; j < HH; j += 32) acc += hr[j] * Wp[j];
    for (int off = 16; off > 0; off >>= 1) acc += __shfl_down(acc, off, 32);
    if (lane == 0) logits[warp] = acc + bp[0];
}

__global__ void seg_softmax_stats(const float* __restrict__ logits, const int* __restrict__ batch,
                                  float* __restrict__ mbuf, float* __restrict__ sbuf, int n) {
    const int b = blockIdx.x;
    __shared__ float red[256];
    const int tid = threadIdx.x;
    float mx = -3.0e38f;
    for (int i = tid; i < n; i += 256)
        if (batch[i] == b) mx = fmaxf(mx, logits[i]);
    red[tid] = mx; __syncthreads();
    for (int st = 128; st > 0; st >>= 1) {
        if (tid < st) red[tid] = fmaxf(red[tid], red[tid + st]);
        __syncthreads();
    }
    const float m = red[0]; __syncthreads();
    float sm = 0.0f;
    for (int i = tid; i < n; i += 256)
        if (batch[i] == b) sm += __expf(logits[i] - m);
    red[tid] = sm; __syncthreads();
    for (int st = 128; st > 0; st >>= 1) {
        if (tid < st) red[tid] += red[tid + st];
        __syncthreads();
    }
    if (tid == 0) { mbuf[b] = m; sbuf[b] = red[0]; }
}

__global__ void softmax_apply(const float* __restrict__ logits, const int* __restrict__ batch,
                              const float* __restrict__ mbuf, const float* __restrict__ sbuf,
                              float* __restrict__ p, int n) {
    const int i = blockIdx.x * blockDim.x + threadIdx.x;
    if (i >= n) return;
    const int b = batch[i];
    p[i] = __expf(logits[i] - mbuf[b]) / (sbuf[b] + 1e-16f);
}

#define GCHUNK 128
__global__ void g_kernel(const float* __restrict__ p, const float* __restrict__ h,
                         const int* __restrict__ batch, float* __restrict__ g, int n) {
    const int j  = threadIdx.x;                 // 0..127
    const int n0 = blockIdx.x * GCHUNK;
    if (n0 >= n) return;
    const int n1 = min(n0 + GCHUNK, n);
    int curb = batch[n0];
    float acc = 0.0f;
    for (int i = n0; i < n1; ++i) {
        const int b = batch[i];                 // sorted -> at most a couple of flushes
        if (b != curb) {
            atomicAdd(&g[curb * HH + j], acc);
            acc = 0.0f; curb = b;
        }
        acc += p[i] * h[(size_t)i * HH + j];
    }
    atomicAdd(&g[curb * HH + j], acc);
}

__global__ void refine_kernel(float* __restrict__ instr, const float* __restrict__ g,
                              const float* __restrict__ Wr, const float* __restrict__ Wg) {
    const int bk = blockIdx.x;     // b*K + k
    const int j  = threadIdx.x;    // 0..127
    __shared__ float ctx[4 * HH];
    const int b  = bk / KK;
    const float iv = instr[(size_t)bk * HH + j];
    const float gv = g[b * HH + j];
    ctx[j]          = iv;
    ctx[HH + j]     = gv;
    ctx[2 * HH + j] = iv * gv;
    ctx[3 * HH + j] = iv - gv;
    __syncthreads();
    float ac = 0.0f, ag = 0.0f;
    for (int i = 0; i < 4 * HH; ++i) {
        const float c = ctx[i];
        ac += c * Wr[(size_t)i * HH + j];
        ag += c * Wg[(size_t)i * HH + j];
    }
    const float cand = tanhf(ac);
    const float gate = 1.0f / (1.0f + __expf(-ag));
    instr[(size_t)bk * HH + j] = gate * cand + (1.0f - gate) * iv;
}

// =================================================================
// Host orchestration
// =================================================================
extern "C" void kernel_launch(void* const* d_in, const int* in_sizes, int n_in,
                              void* d_out, int out_size, void* d_ws, size_t ws_size,
                              hipStream_t stream) {
    const float* question_tokens = (const float*)d_in[0];
    const float* question_mask   = (const float*)d_in[1];
    const float* x               = (const float*)d_in[2];
    const int*   edge_index      = (const int*)d_in[3];
    const int*   edge_type       = (const int*)d_in[4];
    const float* edge_attr       = (const float*)d_in[5];
    const int*   seed_mask       = (const int*)d_in[6];
    const int*   batch           = (const int*)d_in[7];
    const float* W_node = (const float*)d_in[8];
    const float* b_node = (const float*)d_in[9];
    const float* W_q    = (const float*)d_in[10];
    const float* b_q    = (const float*)d_in[11];
    const float* instr_init = (const float*)d_in[12];
    const float* W_tok  = (const float*)d_in[13];
    const float* b_tok  = (const float*)d_in[14];
    const float* Wk     = (const float*)d_in[15];
    const float* bk     = (const float*)d_in[16];
    const float* We     = (const float*)d_in[17];
    const float* be     = (const float*)d_in[18];
    const float* Wf     = (const float*)d_in[19];
    const float* bf     = (const float*)d_in[20];
    const float* W_reset = (const float*)d_in[21];
    const float* W_gate  = (const float*)d_in[22];
    const float* W_prob  = (const float*)d_in[23];
    const float* b_prob  = (const float*)d_in[24];

    const int E = in_sizes[3] / 2;           // 100000
    const int N = in_sizes[6];               // 20000
    const int* src = edge_index;
    const int* dst = edge_index + E;

    // ---- workspace layout (floats) ----
    float* ws = (float*)d_ws;
    float* h      = ws;                     ws += (size_t)N * HH;
    float* h2     = ws;                     ws += (size_t)N * HH;
    float* qt     = ws;                     ws += (size_t)BB * SS * HH;
    float* qr     = ws;                     ws += (size_t)BB * HH;
    float* instr  = ws;                     ws += (size_t)BB * KK * HH;
    float* ep     = ws;                     ws += (size_t)E * HH;
    float* agg    = ws;                     ws += (size_t)N * 2 * KK * HH;
    float* logits = ws;                     ws += (size_t)N;
    float* p      = ws;                     ws += (size_t)N;
    float* mbuf   = ws;                     ws += BB;
    float* sbuf   = ws;                     ws += BB;
    float* gbuf   = ws;                     ws += (size_t)BB * HH;

    // ---- h = relu(x @ W_node + b_node) ----
    gemm_wmma_f32<<<(N + 63) / 64, 256, 0, stream>>>(
        x, NODE_IN, NODE_IN, x, NODE_IN, W_node, b_node, h, N, NODE_IN, 1);

    // ---- qt = relu(question_tokens @ W_q + b_q) ----
    gemm_wmma_f32<<<(BB * SS + 63) / 64, 256, 0, stream>>>(
        question_tokens, QDIM, QDIM, question_tokens, QDIM, W_q, b_q, qt, BB * SS, QDIM, 1);

    q_repr_kernel<<<BB, HH, 0, stream>>>(qt, question_mask, qr);
    instr_gen_kernel<<<BB, HH, 0, stream>>>(qt, question_mask, qr, instr_init,
                                            Wk, bk, W_tok, b_tok, instr);

    float* hcur = h;
    float* hnxt = h2;

    for (int t = 0; t < NITER; ++t) {
        seed_kernel<<<(N + 255) / 256, 256, 0, stream>>>(seed_mask, p, N);
        for (int l = 0; l < LL; ++l) {
            // ep = edge_attr @ We[l] + be[l]   (no relu)
            gemm_wmma_f32<<<(E + 63) / 64, 256, 0, stream>>>(
                edge_attr, EDGE_DIM, EDGE_DIM, edge_attr, EDGE_DIM,
                We + (size_t)l * EDGE_DIM * HH, be + (size_t)l * HH, ep, E, EDGE_DIM, 0);

            hipMemsetAsync(agg, 0, (size_t)N * 2 * KK * HH * sizeof(float), stream);
            edge_message_scatter<<<E, HH, 0, stream>>>(src, dst, edge_type, batch,
                                                       ep, instr, p, agg, E);

            // h' = relu(concat([h, agg]) @ Wf[l] + bf[l])
            gemm_wmma_f32<<<(N + 63) / 64, 256, 0, stream>>>(
                hcur, HH, HH, agg, 2 * KK * HH,
                Wf + (size_t)l * (2 * KK + 1) * HH * HH, bf + (size_t)l * HH,
                hnxt, N, (2 * KK + 1) * HH, 1);
            { float* tmp = hcur; hcur = hnxt; hnxt = tmp; }

            logits_kernel<<<(N + 3) / 4, 128, 0, stream>>>(hcur, W_prob, b_prob, logits, N);
            seg_softmax_stats<<<BB, 256, 0, stream>>>(logits, batch, mbuf, sbuf, N);
            float* ptgt = (t == NITER - 1 && l == LL - 1) ? (float*)d_out : p;
            softmax_apply<<<(N + 255) / 256, 256, 0, stream>>>(logits, batch, mbuf, sbuf, ptgt, N);
        }
        if (t < NITER - 1) {
            hipMemsetAsync(gbuf, 0, (size_t)BB * HH * sizeof(float), stream);
            g_kernel<<<(N + GCHUNK - 1) / GCHUNK, HH, 0, stream>>>(p, hcur, batch, gbuf, N);
            refine_kernel<<<BB * KK, HH, 0, stream>>>(instr, gbuf, W_reset, W_gate);
        }
    }
}